// SEALGCN_21981642621452
// MI455X (gfx1250) — compile-verified
//
#include <hip/hip_runtime.h>

typedef __attribute__((ext_vector_type(2))) float v2f;
typedef __attribute__((ext_vector_type(8))) float v8f;

#define HDIM   128
#define NGRAPH 1024
#define SXP    132   // LDS pitch for X tile (bank-conflict-free frag reads)
#define SWP    132   // LDS pitch for W^T

// ---------------------------------------------------------------- utilities
__global__ void k_fill(float* __restrict__ p, float v, int n) {
    int i = blockIdx.x * blockDim.x + threadIdx.x;
    if (i < n) p[i] = v;
}

__global__ void k_deg_count(const int* __restrict__ dst, float* __restrict__ deg, int E) {
    int i = blockIdx.x * blockDim.x + threadIdx.x;
    if (i < E) atomicAdd(&deg[dst[i]], 1.0f);
}

__global__ void k_rsqrt_inplace(float* __restrict__ p, int n) {
    int i = blockIdx.x * blockDim.x + threadIdx.x;
    if (i < n) p[i] = rsqrtf(p[i]);
}

// ------------------------------------------------- fp32 WMMA GEMM: Y = X @ W
// X:[nrows,128] row-major, W:[128,128] row-major, Y:[nrows,128]
// Block = 256 threads = 8 waves; block computes a 16x128 strip of Y.
// Wave w owns the 16x16 tile at columns [16w,16w+16), accumulating K in
// steps of 4 with V_WMMA_F32_16X16X4_F32.
__global__ __launch_bounds__(256) void k_gemm_wmma(const float* __restrict__ X,
                                                   const float* __restrict__ W,
                                                   float* __restrict__ Y,
                                                   int nrows) {
    __shared__ float sW[HDIM * SWP];   // W transposed: sW[col*SWP + k]
    __shared__ float sX[16 * SXP];     // 16 rows of X

    const int tid  = threadIdx.x;
    const int lane = tid & 31;
    const int wave = tid >> 5;         // 0..7 -> output column tile
    const int row0 = blockIdx.x * 16;

    // Stage W^T in LDS (reused by all 8 waves / whole K loop).
    for (int i = tid; i < HDIM * HDIM; i += 256) {
        int k = i >> 7;            // W row (K index)
        int j = i & (HDIM - 1);    // W col
        sW[j * SWP + k] = W[i];
    }
    // Stage 16xK strip of X (shared by all 8 waves).
    for (int i = tid; i < 16 * HDIM; i += 256) {
        int r = i >> 7, c = i & (HDIM - 1);
        int gr = row0 + r;
        sX[r * SXP + c] = (gr < nrows) ? X[(size_t)gr * HDIM + c] : 0.0f;
    }
    __syncthreads();

    const int half = lane >> 4;        // 0: K+{0,1}, 1: K+{2,3}
    const int l16  = lane & 15;
    const float* aRow = &sX[l16 * SXP + 2 * half];
    const float* bCol = &sW[(wave * 16 + l16) * SWP + 2 * half];

    v8f acc = {};
#pragma unroll
    for (int k = 0; k < HDIM; k += 4) {
        v2f a, b;
        a.x = aRow[k];  a.y = aRow[k + 1];
        b.x = bCol[k];  b.y = bCol[k + 1];
        acc = __builtin_amdgcn_wmma_f32_16x16x4_f32(false, a, false, b,
                                                    (short)0, acc, false, false);
    }

    // D layout: acc[v] -> row = row0 + v + 8*half, col = wave*16 + l16
#pragma unroll
    for (int v = 0; v < 8; ++v) {
        int r = row0 + v + 8 * half;
        if (r < nrows) Y[(size_t)r * HDIM + wave * 16 + l16] = acc[v];
    }
}

// ------------------------------------------- edge scatter: one wave per edge
// Work items [0,E): real edges; [E,E+N): self loops (norm = dinv^2).
__global__ __launch_bounds__(256) void k_agg(const float* __restrict__ XW,
                                             const int* __restrict__ src,
                                             const int* __restrict__ dst,
                                             const float* __restrict__ dinv,
                                             float* __restrict__ OUT,
                                             int E, int N) {
    int gw   = (int)((blockIdx.x * (size_t)blockDim.x + threadIdx.x) >> 5);
    int lane = threadIdx.x & 31;
    if (gw >= E + N) return;

    int s, d; float norm;
    if (gw < E) {
        s = src[gw]; d = dst[gw];
        norm = dinv[s] * dinv[d];
    } else {
        s = d = gw - E;
        float di = dinv[s];
        norm = di * di;
    }
    float4 v = ((const float4*)(XW + (size_t)s * HDIM))[lane];  // 16B/lane = 512B/row
    float* q = OUT + (size_t)d * HDIM + lane * 4;
    atomicAdd(q + 0, v.x * norm);
    atomicAdd(q + 1, v.y * norm);
    atomicAdd(q + 2, v.z * norm);
    atomicAdd(q + 3, v.w * norm);
}

// ------------------------------------------------ fused bias + BN(+ReLU)
__global__ void k_bn(const float* __restrict__ AGG, const float* __restrict__ bias,
                     const float* __restrict__ gamma, const float* __restrict__ beta,
                     const float* __restrict__ mean, const float* __restrict__ var,
                     float* __restrict__ H, int total, int do_relu) {
    int i = blockIdx.x * blockDim.x + threadIdx.x;
    if (i >= total) return;
    int c = i & (HDIM - 1);
    float x = AGG[i] + bias[c];
    float y = (x - mean[c]) * rsqrtf(var[c] + 1e-5f) * gamma[c] + beta[c];
    if (do_relu) y = fmaxf(y, 0.0f);
    H[i] = y;
}

// ------------------------------------------------ mean-pool: one wave per node
__global__ __launch_bounds__(256) void k_pool(const float* __restrict__ H,
                                              const int* __restrict__ batch,
                                              float* __restrict__ SUM,
                                              float* __restrict__ CNT, int N) {
    int gw   = (int)((blockIdx.x * (size_t)blockDim.x + threadIdx.x) >> 5);
    int lane = threadIdx.x & 31;
    if (gw >= N) return;
    int g = batch[gw];
    float4 v = ((const float4*)(H + (size_t)gw * HDIM))[lane];
    float* q = SUM + (size_t)g * HDIM + lane * 4;
    atomicAdd(q + 0, v.x);
    atomicAdd(q + 1, v.y);
    atomicAdd(q + 2, v.z);
    atomicAdd(q + 3, v.w);
    if (lane == 0) atomicAdd(&CNT[g], 1.0f);
}

// ------------------------------------------------ classifier: block per graph
__global__ __launch_bounds__(64) void k_cls(const float* __restrict__ SUM,
                                            const float* __restrict__ CNT,
                                            const float* __restrict__ W1,
                                            const float* __restrict__ b1,
                                            const float* __restrict__ W2,
                                            const float* __restrict__ b2,
                                            float* __restrict__ OUT) {
    __shared__ float sp[HDIM];
    __shared__ float red[64];
    int g = blockIdx.x, t = threadIdx.x;   // t in [0,64)
    float inv = 1.0f / fmaxf(CNT[g], 1.0f);
    sp[t]      = SUM[g * HDIM + t]      * inv;
    sp[t + 64] = SUM[g * HDIM + t + 64] * inv;
    __syncthreads();
    float z = b1[t];
#pragma unroll 8
    for (int k = 0; k < HDIM; ++k) z += sp[k] * W1[k * 64 + t];
    z = fmaxf(z, 0.0f);
    red[t] = z * W2[t];
    __syncthreads();
    for (int s2 = 32; s2 > 0; s2 >>= 1) {
        if (t < s2) red[t] += red[t + s2];
        __syncthreads();
    }
    if (t == 0) OUT[g] = red[0] + b2[0];
}

// ---------------------------------------------------------------- launcher
extern "C" void kernel_launch(void* const* d_in, const int* in_sizes, int n_in,
                              void* d_out, int out_size, void* d_ws, size_t ws_size,
                              hipStream_t stream) {
    const float* x     = (const float*)d_in[0];
    const int*   ei    = (const int*)d_in[1];
    const int*   batch = (const int*)d_in[2];
    const float* Ws    = (const float*)d_in[3];
    const float* bs    = (const float*)d_in[4];
    const float* gam   = (const float*)d_in[5];
    const float* bet   = (const float*)d_in[6];
    const float* mu    = (const float*)d_in[7];
    const float* var   = (const float*)d_in[8];
    const float* W1    = (const float*)d_in[9];
    const float* b1    = (const float*)d_in[10];
    const float* W2    = (const float*)d_in[11];
    const float* b2    = (const float*)d_in[12];
    float*       out   = (float*)d_out;

    const int N = in_sizes[0] / HDIM;     // 50000
    const int E = in_sizes[1] / 2;        // 800000
    const int* src = ei;
    const int* dst = ei + E;

    float* ws   = (float*)d_ws;
    float* buf0 = ws;                                // h        [N,128]
    float* buf1 = buf0 + (size_t)N * HDIM;           // xw       [N,128]
    float* buf2 = buf1 + (size_t)N * HDIM;           // agg      [N,128]
    float* dinv = buf2 + (size_t)N * HDIM;           // deg/dinv [N]
    float* SUM  = dinv + N;                          // [1024,128]
    float* CNT  = SUM + (size_t)NGRAPH * HDIM;       // [1024]

    const int T = 256;
    // degrees with self loops -> dinv (layer invariant)
    k_fill<<<(N + T - 1) / T, T, 0, stream>>>(dinv, 1.0f, N);
    k_deg_count<<<(E + T - 1) / T, T, 0, stream>>>(dst, dinv, E);
    k_rsqrt_inplace<<<(N + T - 1) / T, T, 0, stream>>>(dinv, N);

    const int nWavesAgg = E + N;
    for (int layer = 0; layer < 3; ++layer) {
        const float* hin = (layer == 0) ? x : buf0;
        k_gemm_wmma<<<(N + 15) / 16, 256, 0, stream>>>(hin, Ws + (size_t)layer * HDIM * HDIM,
                                                       buf1, N);
        k_fill<<<(N * HDIM + T - 1) / T, T, 0, stream>>>(buf2, 0.0f, N * HDIM);
        k_agg<<<((size_t)nWavesAgg * 32 + T - 1) / T, T, 0, stream>>>(buf1, src, dst, dinv,
                                                                      buf2, E, N);
        k_bn<<<(N * HDIM + T - 1) / T, T, 0, stream>>>(buf2, bs + layer * HDIM,
                                                       gam + layer * HDIM, bet + layer * HDIM,
                                                       mu + layer * HDIM, var + layer * HDIM,
                                                       buf0, N * HDIM, layer < 2 ? 1 : 0);
    }

    k_fill<<<(NGRAPH * HDIM + T - 1) / T, T, 0, stream>>>(SUM, 0.0f, NGRAPH * HDIM);
    k_fill<<<(NGRAPH + T - 1) / T, T, 0, stream>>>(CNT, 0.0f, NGRAPH);
    k_pool<<<((size_t)N * 32 + T - 1) / T, T, 0, stream>>>(buf0, batch, SUM, CNT, N);
    k_cls<<<NGRAPH, 64, 0, stream>>>(SUM, CNT, W1, b1, W2, b2, out);
}